// MessagePassingBonded_24988119728557
// MI455X (gfx1250) — compile-verified
//
#include <hip/hip_runtime.h>
#include <math.h>

#define N_NODES 500000
#define N_EDGES 2000000
#define F_IN    117
#define F_PAD   124      // K padded to mult of 4; stride 124 -> conflict-free LDS
#define HID     128
#define HS      132      // LDS row stride for 128-wide tiles (bank-conflict-free)
#define ROWS    32       // rows per block (two 16-row WMMA tiles per wave)
#define NLAYERS 3

typedef __attribute__((ext_vector_type(2))) float v2f;
typedef __attribute__((ext_vector_type(8))) float v8f;

__device__ __forceinline__ v8f wmma_f32_4(v2f a, v2f b, v8f c) {
    // D = A(16x4,f32) * B(4x16,f32) + C(16x16,f32); full fp32 matrix op on CDNA5
    return __builtin_amdgcn_wmma_f32_16x16x4_f32(
        /*neg_a=*/false, a, /*neg_b=*/false, b,
        /*c_mod=*/(short)0, c, /*reuse_a=*/false, /*reuse_b=*/false);
}

// ---------------------------------------------------------------- utilities
__global__ void zero_f4(float4* __restrict__ p, int n4) {
    int i = blockIdx.x * blockDim.x + threadIdx.x;
    if (i < n4) p[i] = make_float4(0.f, 0.f, 0.f, 0.f);
}

__global__ void deg_kernel(const int* __restrict__ edst, float* __restrict__ deg) {
    int e = blockIdx.x * blockDim.x + threadIdx.x;
    if (e < N_EDGES) unsafeAtomicAdd(&deg[edst[e]], 1.0f);
}

__global__ void invdeg_kernel(float* __restrict__ deg) {
    int i = blockIdx.x * blockDim.x + threadIdx.x;
    if (i < N_NODES) deg[i] = 1.0f / fmaxf(deg[i], 1.0f);
}

// One wave per edge: lane carries 4 of the 128 channels.
__global__ void scatter_kernel(const float* __restrict__ h,
                               const int* __restrict__ esrc,
                               const int* __restrict__ edst,
                               float* __restrict__ neigh) {
    int wave = (int)((blockIdx.x * blockDim.x + threadIdx.x) >> 5);
    int lane = threadIdx.x & 31;
    if (wave >= N_EDGES) return;
    int s = esrc[wave];
    int d = edst[wave];
    const float4 v = *(const float4*)(h + (size_t)s * HID + lane * 4);
    float* dp = neigh + (size_t)d * HID + lane * 4;
    unsafeAtomicAdd(dp + 0, v.x);
    unsafeAtomicAdd(dp + 1, v.y);
    unsafeAtomicAdd(dp + 2, v.z);
    unsafeAtomicAdd(dp + 3, v.w);
}

// ------------------------------------------- input projection: tanh(x@W_in+b)
// block = 256 threads = 8 waves; block tile = 32 rows x 128 cols.
// Wave w owns cols [16w,16w+16) and TWO 16-row tiles, sharing B fragments.
// K loop split: 29 guard-free steps (k<=115 < 117) + one branchless tail step.
__global__ void __launch_bounds__(256)
proj_kernel(const float* __restrict__ x, const float* __restrict__ W_in,
            const float* __restrict__ b_in, float* __restrict__ h) {
    __shared__ float xs[ROWS * F_PAD];
    const int tid  = threadIdx.x;
    const int row0 = blockIdx.x * ROWS;          // 500000 % 32 == 0

    for (int i = tid; i < ROWS * F_PAD; i += 256) {
        int r = i / F_PAD, c = i - r * F_PAD;
        xs[i] = (c < F_IN) ? x[(size_t)(row0 + r) * F_IN + c] : 0.0f;
    }
    __syncthreads();

    const int wave = tid >> 5, lane = tid & 31;
    const int half = lane >> 4, lrow = lane & 15;
    const int col  = wave * 16 + lrow;

    v8f acc0 = {}, acc1 = {};
    // main loop: kk in [0,116), ka+1 <= 115 < F_IN -> no guards, no divergence
    for (int kk = 0; kk < 116; kk += 4) {
        const int ka = kk + 2 * half;            // A layout: vgpr{0,1}=K{ka,ka+1}
        v2f b;
        b[0] = W_in[(size_t)ka       * HID + col];
        b[1] = W_in[(size_t)(ka + 1) * HID + col];
        v2f a0, a1;
        a0[0] = xs[lrow * F_PAD + ka];
        a0[1] = xs[lrow * F_PAD + ka + 1];
        a1[0] = xs[(16 + lrow) * F_PAD + ka];
        a1[1] = xs[(16 + lrow) * F_PAD + ka + 1];
        acc0 = wmma_f32_4(a0, b, acc0);
        acc1 = wmma_f32_4(a1, b, acc1);
    }
    // tail (kk=116): only k=116 is live; uniform load + select, no exec masking
    {
        const int ka = 116 + 2 * half;
        float w116 = W_in[(size_t)116 * HID + col];  // loaded by all lanes
        v2f b;
        b[0] = half ? 0.0f : w116;                   // v_cndmask, branchless
        b[1] = 0.0f;
        v2f a0, a1;                                  // xs is zero-padded past 117
        a0[0] = xs[lrow * F_PAD + ka];
        a0[1] = xs[lrow * F_PAD + ka + 1];
        a1[0] = xs[(16 + lrow) * F_PAD + ka];
        a1[1] = xs[(16 + lrow) * F_PAD + ka + 1];
        acc0 = wmma_f32_4(a0, b, acc0);
        acc1 = wmma_f32_4(a1, b, acc1);
    }

    const float bias = b_in[col];
    float* hp = h + (size_t)row0 * HID;
#pragma unroll
    for (int r = 0; r < 8; ++r) {                // D: M = r + 8*half, N = col
        int row = r + 8 * half;
        hp[(size_t)row * HID + col]        = tanhf(acc0[r] + bias);
        hp[(size_t)(16 + row) * HID + col] = tanhf(acc1[r] + bias);
    }
}

// ---------------- layer: relu(h@W_self + (neigh*inv_deg)@W_neigh + b) -------
__global__ void __launch_bounds__(256)
layer_kernel(const float* __restrict__ h, const float* __restrict__ neigh,
             const float* __restrict__ invdeg,
             const float* __restrict__ Wself, const float* __restrict__ Wneigh,
             const float* __restrict__ bias, float* __restrict__ out) {
    __shared__ float hs[ROWS * HS];
    __shared__ float ns[ROWS * HS];
    const int tid  = threadIdx.x;
    const int row0 = blockIdx.x * ROWS;

    for (int i = tid; i < ROWS * (HID / 4); i += 256) { // 1024 float4 chunks
        int r  = i >> 5;                                // 32 float4 per row
        int c4 = (i & 31) * 4;
        float4 hv = *(const float4*)(h + (size_t)(row0 + r) * HID + c4);
        *(float4*)(&hs[r * HS + c4]) = hv;
        float inv = invdeg[row0 + r];
        float4 nv = *(const float4*)(neigh + (size_t)(row0 + r) * HID + c4);
        nv.x *= inv; nv.y *= inv; nv.z *= inv; nv.w *= inv;
        *(float4*)(&ns[r * HS + c4]) = nv;
    }
    __syncthreads();

    const int wave = tid >> 5, lane = tid & 31;
    const int half = lane >> 4, lrow = lane & 15;
    const int col  = wave * 16 + lrow;

    v8f acc0 = {}, acc1 = {};
    for (int kk = 0; kk < HID; kk += 4) {
        const int ka = kk + 2 * half;
        v2f bs, bn, a;
        bs[0] = Wself[(size_t)ka       * HID + col];
        bs[1] = Wself[(size_t)(ka + 1) * HID + col];
        bn[0] = Wneigh[(size_t)ka       * HID + col];
        bn[1] = Wneigh[(size_t)(ka + 1) * HID + col];

        a[0] = hs[lrow * HS + ka];
        a[1] = hs[lrow * HS + ka + 1];
        acc0 = wmma_f32_4(a, bs, acc0);
        a[0] = hs[(16 + lrow) * HS + ka];
        a[1] = hs[(16 + lrow) * HS + ka + 1];
        acc1 = wmma_f32_4(a, bs, acc1);

        a[0] = ns[lrow * HS + ka];
        a[1] = ns[lrow * HS + ka + 1];
        acc0 = wmma_f32_4(a, bn, acc0);
        a[0] = ns[(16 + lrow) * HS + ka];
        a[1] = ns[(16 + lrow) * HS + ka + 1];
        acc1 = wmma_f32_4(a, bn, acc1);
    }

    const float bv = bias[col];
#pragma unroll
    for (int r = 0; r < 8; ++r) {
        int row = r + 8 * half;
        out[(size_t)(row0 + row) * HID + col]      = fmaxf(acc0[r] + bv, 0.0f);
        out[(size_t)(row0 + 16 + row) * HID + col] = fmaxf(acc1[r] + bv, 0.0f);
    }
}

// ---------------------------------------------------------------------------
extern "C" void kernel_launch(void* const* d_in, const int* in_sizes, int n_in,
                              void* d_out, int out_size, void* d_ws, size_t ws_size,
                              hipStream_t stream) {
    const float* x        = (const float*)d_in[0];
    const int*   esrc     = (const int*)  d_in[1];
    const int*   edst     = (const int*)  d_in[2];
    const float* W_in     = (const float*)d_in[3];
    const float* b_in     = (const float*)d_in[4];
    const float* W_self   = (const float*)d_in[5];
    const float* W_neigh  = (const float*)d_in[6];
    const float* b_layers = (const float*)d_in[7];
    float*       out      = (float*)d_out;

    // ws layout: hA (N*H f32) | neigh (N*H f32) | invdeg (N f32)  => ~514 MB
    float* hA     = (float*)d_ws;
    float* neigh  = hA    + (size_t)N_NODES * HID;
    float* invdeg = neigh + (size_t)N_NODES * HID;

    const int NH4 = N_NODES * HID / 4;   // 16M float4

    // degree -> inverse degree (recomputed every call; deterministic work)
    zero_f4<<<(N_NODES / 4 + 255) / 256, 256, 0, stream>>>((float4*)invdeg, N_NODES / 4);
    deg_kernel<<<(N_EDGES + 255) / 256, 256, 0, stream>>>(edst, invdeg);
    invdeg_kernel<<<(N_NODES + 255) / 256, 256, 0, stream>>>(invdeg);

    // input projection
    proj_kernel<<<N_NODES / ROWS, 256, 0, stream>>>(x, W_in, b_in, hA);

    // layer ping-pong: hA -> out -> hA -> out (final result lands in d_out)
    const float* hins[NLAYERS]  = { hA, out, hA };
    float*       houts[NLAYERS] = { out, hA, out };
    for (int l = 0; l < NLAYERS; ++l) {
        zero_f4<<<(NH4 + 255) / 256, 256, 0, stream>>>((float4*)neigh, NH4);
        scatter_kernel<<<N_EDGES / 8, 256, 0, stream>>>(hins[l], esrc, edst, neigh);
        layer_kernel<<<N_NODES / ROWS, 256, 0, stream>>>(
            hins[l], neigh, invdeg,
            W_self  + (size_t)l * HID * HID,
            W_neigh + (size_t)l * HID * HID,
            b_layers + (size_t)l * HID,
            houts[l]);
    }
}